// QuanvolutionFilter_65481071408606
// MI455X (gfx1250) — compile-verified
//
#include <hip/hip_runtime.h>
#include <cstdint>

// Quanvolution filter, algebraically reduced:
//   out[b, 4p+k] = prod_{j<=k} cos(x_patch[j] + params[0,j])
// (RZ params drop out of |amp|^2; same-axis RY's compose; the CNOT chain turns
//  Z_k into the parity b0^..^bk, whose product-state expectation is the
//  cumulative cosine product.)
//
// Memory-bound streaming kernel: ~25.7 MB total traffic -> ~1.1 us floor at
// 23.3 TB/s HBM. Data movement uses BOTH gfx1250 DMA paths on disjoint halves
// of the block tile:
//   - wave 0: TENSOR_LOAD_TO_LDS (TDM descriptor, TENSORcnt) for images 0..1
//   - all waves: global_load_async_to_lds_b128 (ASYNCcnt) for images 2..3
// then ds_load_b128 reads and b128 coalesced stores.

#define THREADS 256
#define IMGS_PER_BLOCK 4
#define FLOATS_PER_IMG 784               // 28x28
#define TILE_FLOATS (IMGS_PER_BLOCK * FLOATS_PER_IMG)   // 3136
#define TILE_VEC4   (TILE_FLOATS / 4)                   // 784 float4s
#define TDM_FLOATS  (TILE_FLOATS / 2)                   // 1568 floats via TDM
#define TDM_VEC4    (TDM_FLOATS / 4)                    // 392 float4s
#define TASKS_PER_BLOCK (IMGS_PER_BLOCK * 98)           // 392

typedef __attribute__((ext_vector_type(4))) unsigned int u32x4;
typedef __attribute__((ext_vector_type(8))) unsigned int u32x8;

__global__ __launch_bounds__(THREADS)
void quanv_kernel(const float* __restrict__ x,
                  const float* __restrict__ params,
                  float* __restrict__ out)
{
    __shared__ float tile[TILE_FLOATS];

    const int t = threadIdx.x;
    __builtin_assume(t < THREADS);
    const int blk = blockIdx.x;

    const float* gbase = x + (size_t)blk * TILE_FLOATS;

    // LDS byte offset of tile[0]: generic shared address low 32 bits are the
    // LDS offset (CDNA5 ISA 10.2: shared aperture occupies addr[63:32]).
    const uint32_t lds_base = (uint32_t)(uintptr_t)(&tile[0]);

    // --- Stage 1a: TDM descriptor copy of first half (wave 0 only) ---------
    if (t < 32) {   // wave-uniform branch; TDM issues once per wave, EXEC-ignored
        const uint64_t ga = (uint64_t)(uintptr_t)gbase;
        u32x4 g0;
        g0.x = 1u;                                   // count=1, user descriptor
        g0.y = lds_base;                             // lds_addr (bytes)
        g0.z = (uint32_t)ga;                         // global_addr[31:0]
        g0.w = (uint32_t)(ga >> 32) | (2u << 30);    // global_addr[56:32], type=2

        u32x8 g1;
        g1[0] = (2u << 16);                          // data_size=2 (4B); no mcast/pad/iter
        g1[1] = ((unsigned)TDM_FLOATS & 0xFFFFu) << 16;  // tensor_dim0[15:0] @ bits 63:48
        g1[2] = ((unsigned)TDM_FLOATS >> 16) | (1u << 16); // tensor_dim0[31:16], tensor_dim1=1
        g1[3] = ((unsigned)TDM_FLOATS) << 16;        // tile_dim0 = 1568 @ bits 127:112
        g1[4] = 1u;                                  // tile_dim1 = 1, tile_dim2 = 0
        g1[5] = (unsigned)TDM_FLOATS;                // tensor_dim0_stride[31:0]
        g1[6] = 0u;
        g1[7] = 0u;

        asm volatile("tensor_load_to_lds %0, %1"
                     :
                     : "s"(g0), "s"(g1)
                     : "memory");
    }

    // --- Stage 1b: async b128 lane copies of second half (all waves) -------
#pragma unroll
    for (int k = 0; k < 2; ++k) {
        int i = TDM_VEC4 + t + k * THREADS;          // float4 index 392..783
        if (i < TILE_VEC4) {
            uint32_t voff = (uint32_t)i * 16u;       // global byte offset (GVS mode)
            uint32_t ldso = lds_base + voff;         // LDS byte address
            asm volatile("global_load_async_to_lds_b128 %0, %1, %2"
                         :
                         : "v"(ldso), "v"(voff), "s"(gbase)
                         : "memory");
        }
    }

    // Per-wave completion, then workgroup-wide visibility.
    if (t < 32) {
        __builtin_amdgcn_s_wait_tensorcnt(0);        // s_wait_tensorcnt 0
    }
    asm volatile("s_wait_asynccnt 0" ::: "memory");
    __syncthreads();

    // Only params[0,0..3] matter (RZ params are phase-only).
    const float q0 = params[0];
    const float q1 = params[1];
    const float q2 = params[2];
    const float q3 = params[3];

    // --- Stage 2: compute 2 adjacent 2x2 patches per task ------------------
#pragma unroll
    for (int k = 0; k < 2; ++k) {
        int j = t + k * THREADS;
        if (j < TASKS_PER_BLOCK) {
            int img = j / 98;
            int rem = j - img * 98;
            int r   = rem / 7;              // row-pair 0..13
            int pc  = rem - r * 7;          // column-pair 0..6 (cols 4pc..4pc+3)

            // row 2r starts at r*56 floats, row 2r+1 at +28; both 16B-aligned.
            const int base = img * FLOATS_PER_IMG + r * 56 + pc * 4;
            float4 topv = *(const float4*)&tile[base];       // ds_load_b128
            float4 botv = *(const float4*)&tile[base + 28];  // ds_load_b128

            // patch A = (top.x, top.y, bot.x, bot.y)
            float e0a = __cosf(topv.x + q0);
            float e1a = e0a * __cosf(topv.y + q1);
            float e2a = e1a * __cosf(botv.x + q2);
            float e3a = e2a * __cosf(botv.y + q3);
            // patch B = (top.z, top.w, bot.z, bot.w)
            float e0b = __cosf(topv.z + q0);
            float e1b = e0b * __cosf(topv.w + q1);
            float e2b = e1b * __cosf(botv.z + q2);
            float e3b = e2b * __cosf(botv.w + q3);

            // patches r*14+2pc and r*14+2pc+1 -> 8 contiguous output floats
            float* obase = out + (size_t)(blk * IMGS_PER_BLOCK + img) * FLOATS_PER_IMG
                               + r * 56 + pc * 8;
            *(float4*)(obase)     = make_float4(e0a, e1a, e2a, e3a);
            *(float4*)(obase + 4) = make_float4(e0b, e1b, e2b, e3b);
        }
    }
}

extern "C" void kernel_launch(void* const* d_in, const int* in_sizes, int n_in,
                              void* d_out, int out_size, void* d_ws, size_t ws_size,
                              hipStream_t stream) {
    const float* x      = (const float*)d_in[0];   // (B, 784) float32
    const float* params = (const float*)d_in[1];   // (2, 4)  float32
    float*       out    = (float*)d_out;           // (B, 784) float32

    int nimg   = in_sizes[0] / FLOATS_PER_IMG;                  // 4096
    int blocks = (nimg + IMGS_PER_BLOCK - 1) / IMGS_PER_BLOCK;  // 1024

    quanv_kernel<<<blocks, THREADS, 0, stream>>>(x, params, out);
}